// Diffeo_50818053046994
// MI455X (gfx1250) — compile-verified
//
#include <hip/hip_runtime.h>
#include <math.h>

typedef __attribute__((ext_vector_type(2))) float v2f;
typedef __attribute__((ext_vector_type(4))) float v4f;
typedef __attribute__((ext_vector_type(8))) float v8f;

#define CUT   16
#define NPIX  512           // h == w == 512
#define NBC   (64 * 3)      // batch * channels

// ---------------------------------------------------------------------------
// Kernel 1: sin table S[512][16] and M[512][16] = S @ (c_u * e)
//   S[p][j] = sin(pi * p/511 * (j+1))
//   M[p][j] = sum_i c_u[i][j] * e(i,j) * sin(pi * p/511 * (i+1))
//   e(i,j)  = (r < 16.5) / r,  r = sqrt((i+1)^2 + (j+1)^2)
// 8192 threads, trivial cost.
// ---------------------------------------------------------------------------
__global__ __launch_bounds__(256) void diffeo_field_stage1(
    const float* __restrict__ c_u,
    float* __restrict__ S,
    float* __restrict__ M)
{
    int t = blockIdx.x * blockDim.x + threadIdx.x;   // 0..8191
    int p = t >> 4;                                   // pixel 0..511
    int j = t & 15;                                   // mode  0..15
    const float PI = 3.14159265358979323846f;
    float xp = (float)p / (float)(NPIX - 1);

    S[t] = sinf(PI * xp * (float)(j + 1));

    float kj = (float)(j + 1);
    float acc = 0.0f;
    #pragma unroll
    for (int i = 0; i < CUT; ++i) {
        float ki = (float)(i + 1);
        float r  = sqrtf(ki * ki + kj * kj);
        float e  = (r < (float)CUT + 0.5f) ? (1.0f / r) : 0.0f;
        acc += c_u[i * CUT + j] * e * sinf(PI * xp * ki);
    }
    M[t] = acc;
}

// ---------------------------------------------------------------------------
// Kernel 2: dx[y][x] = dxscale * sum_j S[y][j] * M[x][j]
// One wave32 per 16x16 output tile; K=16 via 4 chained V_WMMA_F32_16X16X4_F32.
//   A (16x4 f32): lane l -> A[l&15, (l<16?0:2)+{0,1}] in 2 VGPRs
//   B (4x16 f32): lane l -> B[(l<16?0:2)+{0,1}, l&15], B[k,n] = M[x0+n][k]
//   D (16x16 f32, 8 VGPRs): D[r] = U[y0 + r + (l<16?0:8), x0 + (l&15)]
// EXEC is all-ones (32-thread blocks, no divergence) as WMMA requires.
// ---------------------------------------------------------------------------
__global__ __launch_bounds__(32) void diffeo_field_wmma(
    const float* __restrict__ S,
    const float* __restrict__ M,
    float* __restrict__ dxf,
    float dxscale)
{
    int tile = blockIdx.x;              // 0..1023
    int ty = tile >> 5, tx = tile & 31;
    int y0 = ty << 4,   x0 = tx << 4;

    int lane  = threadIdx.x;
    int lm    = lane & 15;
    int khalf = (lane >> 4) << 1;       // 0 for lanes 0-15, 2 for lanes 16-31

    v8f acc = {};
    #pragma unroll
    for (int kk = 0; kk < 4; ++kk) {
        int kb = (kk << 2) + khalf;     // global K for this lane's .x slot
        v2f a, b;
        a.x = S[(y0 + lm) * CUT + kb];
        a.y = S[(y0 + lm) * CUT + kb + 1];
        b.x = M[(x0 + lm) * CUT + kb];
        b.y = M[(x0 + lm) * CUT + kb + 1];
        // (neg_a, A, neg_b, B, c_mod, C, reuse_a, reuse_b)
        acc = __builtin_amdgcn_wmma_f32_16x16x4_f32(
            false, a, false, b, (short)0, acc, false, false);
    }

    int row0 = y0 + ((lane >> 4) << 3);
    int col  = x0 + lm;
    #pragma unroll
    for (int r = 0; r < 8; ++r)
        dxf[(row0 + r) * NPIX + col] = dxscale * acc[r];
}

// ---------------------------------------------------------------------------
// Kernel 3: 1-D linear remap along x (dy == 0 since AXIS=(0,)).
// Memory-bound: 4 outputs per thread, vector NT store; dx field (1 MB) is
// L2-resident and reused across all 192 planes.
// ---------------------------------------------------------------------------
__device__ __forceinline__ float remap1(const float* __restrict__ row,
                                        int x, float d)
{
    float xn = fminf(fmaxf((float)x - d, 0.0f), (float)(NPIX - 1));
    float xff = floorf(xn);
    float xv  = xn - xff;
    int xf = (int)xff;
    int xc = (int)ceilf(xn);
    float v0 = row[xf];
    float v1 = row[xc];
    return (1.0f - xv) * v0 + xv * v1;
}

__global__ __launch_bounds__(256) void diffeo_remap(
    const float* __restrict__ img,
    const float* __restrict__ dxf,
    float* __restrict__ out)
{
    // Each thread handles 4 consecutive x. Groups per plane: 512*128 = 2^16.
    long long t = (long long)blockIdx.x * blockDim.x + threadIdx.x;
    int x4 = (int)(t & 127) << 2;            // 0..508
    int y  = (int)((t >> 7) & (NPIX - 1));
    long long plane = (t >> 16) << 18;       // bc * 512*512

    const v4f dv = *(const v4f*)(dxf + (y << 9) + x4);
    const float* __restrict__ row = img + plane + ((long long)y << 9);

    v4f o;
    o.x = remap1(row, x4 + 0, dv.x);
    o.y = remap1(row, x4 + 1, dv.y);
    o.z = remap1(row, x4 + 2, dv.z);
    o.w = remap1(row, x4 + 3, dv.w);

    __builtin_nontemporal_store(o, (v4f*)(out + plane + (y << 9) + x4));
}

// ---------------------------------------------------------------------------
extern "C" void kernel_launch(void* const* d_in, const int* in_sizes, int n_in,
                              void* d_out, int out_size, void* d_ws, size_t ws_size,
                              hipStream_t stream)
{
    const float* img = (const float*)d_in[0];
    const float* c_u = (const float*)d_in[1];
    // d_in[2] (c_v) is unused: AXIS=(0,) makes dy identically zero.
    float* out = (float*)d_out;

    float* S   = (float*)d_ws;            // 512*16 floats
    float* M   = S + NPIX * CUT;          // 512*16 floats
    float* dxf = M + NPIX * CUT;          // 512*512 floats (~1 MB)

    // dx = sqrt(TEMP * 4/(pi^3 * cut^2 * ln(cut))) * u * w
    double Tw = 4.0 / (M_PI * M_PI * M_PI * (double)(CUT * CUT) * log((double)CUT));
    float dxscale = (float)(sqrt(Tw) * (double)NPIX);

    diffeo_field_stage1<<<(NPIX * CUT) / 256, 256, 0, stream>>>(c_u, S, M);
    diffeo_field_wmma<<<(NPIX / 16) * (NPIX / 16), 32, 0, stream>>>(S, M, dxf, dxscale);

    long long total = (long long)NBC * NPIX * NPIX;   // 50331648 elements
    int nthreads_groups = (int)(total / 4 / 256);     // 49152 blocks
    diffeo_remap<<<nthreads_groups, 256, 0, stream>>>(img, dxf, out);
}